// GCN_66228395704559
// MI455X (gfx1250) — compile-verified
//
#include <hip/hip_runtime.h>

#define N_NODES 50000
#define N_EDGES 400000
#define N_GRAPHS 50

typedef float v2f __attribute__((ext_vector_type(2)));
typedef float v8f __attribute__((ext_vector_type(8)));

// ---------------------------------------------------------------- utilities

__global__ void k_fill(float* __restrict__ p, float v, int n) {
    int i = blockIdx.x * blockDim.x + threadIdx.x;
    if (i < n) p[i] = v;
}

// deg accumulation: deg starts at 1.0 (self loop), add 1 per incoming edge (dst)
__global__ void k_deg(const int* __restrict__ dst, float* __restrict__ deg) {
    int e = blockIdx.x * blockDim.x + threadIdx.x;
    if (e < N_EDGES) atomicAdd(&deg[dst[e]], 1.0f);
}

__global__ void k_rsqrt_inplace(float* __restrict__ d) {
    int i = blockIdx.x * blockDim.x + threadIdx.x;
    if (i < N_NODES) d[i] = rsqrtf(d[i]);
}

__global__ void k_edgenorm(const int* __restrict__ src, const int* __restrict__ dst,
                           const float* __restrict__ dinv, float* __restrict__ en) {
    int e = blockIdx.x * blockDim.x + threadIdx.x;
    if (e < N_EDGES) en[e] = dinv[src[e]] * dinv[dst[e]];
}

// ---------------------------------------------------------------- WMMA GEMM
// H[N, NO] = X[N, K] @ W[K, NO]
// One wave computes one 16x16 tile of H, iterating K in steps of 4 using
// V_WMMA_F32_16X16X4_F32. W column panel (K x 16) is staged in LDS once per
// block. N_NODES (50000) is a multiple of 16, K dims are multiples of 4.
//
// ISA lane layouts (CDNA5 ISA):
//   A 16x4 : lanes 0-15 hold row M=lane, {v0,v1} = K{0,1}; lanes 16-31 = K{2,3}
//   B 4x16 : lanes 0-15 hold col N=lane, {v0,v1} = K{0,1}; lanes 16-31 = K{2,3}
//   C 16x16: vgpr r -> lanes 0-15: (M=r, N=lane); lanes 16-31: (M=8+r, N=lane-16)
template <int K, int NO>
__global__ void k_gemm_wmma(const float* __restrict__ X, const float* __restrict__ W,
                            float* __restrict__ H) {
    __shared__ float sW[K * 16];
    const int tid = threadIdx.x;
    const int n0  = blockIdx.y * 16;

    // stage W panel: rows 0..K-1, cols n0..n0+15 (zero-pad past NO)
    for (int i = tid; i < K * 16; i += blockDim.x) {
        int k = i >> 4, c = i & 15;
        int col = n0 + c;
        sW[i] = (col < NO) ? W[k * NO + col] : 0.0f;
    }
    __syncthreads();

    const int wave = tid >> 5;
    const int lane = tid & 31;
    const int m0 = (blockIdx.x * (blockDim.x >> 5) + wave) * 16;
    if (m0 >= N_NODES) return;   // wave-uniform: EXEC stays all-ones for WMMA

    const int lm = lane & 15;    // A: row in tile / B: col in tile
    const int hi = lane >> 4;    // selects K pair {0,1} vs {2,3}

    v8f acc = {};
    const float* xrow = X + (m0 + lm) * K;
    for (int k = 0; k < K; k += 4) {
        const int ka = k + 2 * hi;
        v2f a, b;
        a.x = xrow[ka];
        a.y = xrow[ka + 1];
        b.x = sW[ka * 16 + lm];
        b.y = sW[(ka + 1) * 16 + lm];
        acc = __builtin_amdgcn_wmma_f32_16x16x4_f32(
            /*neg_a=*/false, a, /*neg_b=*/false, b,
            /*c_mod=*/(short)0, acc, /*reuse_a=*/false, /*reuse_b=*/false);
    }

    const int col = n0 + lm;
    if (col < NO) {
        const int mbase = m0 + hi * 8;
#pragma unroll
        for (int r = 0; r < 8; ++r)
            H[(mbase + r) * NO + col] = acc[r];
    }
}

// ---------------------------------------------------------------- scatter-add
// AGG[dst[e], f] += H[src[e], f] * edge_norm[e]
// one thread per (edge, feature); for NO >= 32 a wave shares one edge ->
// src/dst/en loads are broadcast, H reads + atomics are coalesced.
template <int NO>
__global__ void k_scatter(const int* __restrict__ src, const int* __restrict__ dst,
                          const float* __restrict__ en, const float* __restrict__ H,
                          float* __restrict__ AGG) {
    int tid = blockIdx.x * blockDim.x + threadIdx.x;
    if (tid >= N_EDGES * NO) return;
    int e = tid / NO;            // NO is a power of two -> shift
    int f = tid - e * NO;
    int s = src[e], d = dst[e];
    atomicAdd(&AGG[d * NO + f], H[s * NO + f] * en[e]);
}

// ---------------------------------------------------------------- finalize
// OUT = relu(AGG + H * dinv^2 + bias)
template <int NO>
__global__ void k_finalize(const float* __restrict__ AGG, const float* __restrict__ H,
                           const float* __restrict__ dinv, const float* __restrict__ bias,
                           float* __restrict__ OUT) {
    int tid = blockIdx.x * blockDim.x + threadIdx.x;
    if (tid >= N_NODES * NO) return;
    int i = tid / NO;
    int f = tid - i * NO;
    float dv = dinv[i];
    float v = AGG[tid] + H[tid] * (dv * dv) + bias[f];
    OUT[tid] = fmaxf(v, 0.0f);
}

// ---------------------------------------------------------------- pooling

__global__ void k_counts(const int* __restrict__ batch, float* __restrict__ counts) {
    int i = blockIdx.x * blockDim.x + threadIdx.x;
    if (i < N_NODES) atomicAdd(&counts[batch[i]], 1.0f);
}

__global__ void k_poolsum(const int* __restrict__ batch, const float* __restrict__ X,
                          float* __restrict__ sums) {
    int tid = blockIdx.x * blockDim.x + threadIdx.x;
    if (tid >= N_NODES * 512) return;
    int i = tid >> 9;
    int f = tid & 511;
    atomicAdd(&sums[batch[i] * 512 + f], X[tid]);
}

__global__ void k_pooldiv(float* __restrict__ out, const float* __restrict__ counts) {
    int tid = blockIdx.x * blockDim.x + threadIdx.x;
    if (tid >= N_GRAPHS * 512) return;
    out[tid] /= fmaxf(counts[tid >> 9], 1.0f);
}

// ---------------------------------------------------------------- driver

template <int K, int NO>
static void run_layer(const float* X, const float* Wt, const float* bias,
                      float* Hbuf, float* Gbuf, float* Xout,
                      const int* src, const int* dst, const float* en,
                      const float* dinv, hipStream_t stream) {
    constexpr int WAVES_PER_BLOCK = 4;
    dim3 gb((N_NODES / 16 + WAVES_PER_BLOCK - 1) / WAVES_PER_BLOCK, (NO + 15) / 16);
    k_gemm_wmma<K, NO><<<gb, WAVES_PER_BLOCK * 32, 0, stream>>>(X, Wt, Hbuf);

    int na = N_NODES * NO;
    k_fill<<<(na + 255) / 256, 256, 0, stream>>>(Gbuf, 0.0f, na);

    int ne = N_EDGES * NO;
    k_scatter<NO><<<(ne + 255) / 256, 256, 0, stream>>>(src, dst, en, Hbuf, Gbuf);

    k_finalize<NO><<<(na + 255) / 256, 256, 0, stream>>>(Gbuf, Hbuf, dinv, bias, Xout);
}

extern "C" void kernel_launch(void* const* d_in, const int* in_sizes, int n_in,
                              void* d_out, int out_size, void* d_ws, size_t ws_size,
                              hipStream_t stream) {
    const float* x     = (const float*)d_in[0];
    const int*   ei    = (const int*)d_in[1];
    const int*   batch = (const int*)d_in[2];
    const float* W1 = (const float*)d_in[3];  const float* b1 = (const float*)d_in[4];
    const float* W2 = (const float*)d_in[5];  const float* b2 = (const float*)d_in[6];
    const float* W3 = (const float*)d_in[7];  const float* b3 = (const float*)d_in[8];
    const float* W4 = (const float*)d_in[9];  const float* b4 = (const float*)d_in[10];
    const float* W5 = (const float*)d_in[11]; const float* b5 = (const float*)d_in[12];

    const int* src = ei;             // edge_index[0]
    const int* dst = ei + N_EDGES;   // edge_index[1]

    // workspace layout (floats)
    float* ws    = (float*)d_ws;
    float* Xbuf  = ws;                                   // N * 512
    float* Hbuf  = Xbuf + (size_t)N_NODES * 512;         // N * 512
    float* Gbuf  = Hbuf + (size_t)N_NODES * 512;         // N * 512
    float* dinv  = Gbuf + (size_t)N_NODES * 512;         // N
    float* enorm = dinv + N_NODES;                       // E
    float* cnts  = enorm + N_EDGES;                      // B

    // ---- normalization precompute: deg -> dinv -> edge_norm
    k_fill<<<(N_NODES + 255) / 256, 256, 0, stream>>>(dinv, 1.0f, N_NODES);
    k_deg<<<(N_EDGES + 255) / 256, 256, 0, stream>>>(dst, dinv);
    k_rsqrt_inplace<<<(N_NODES + 255) / 256, 256, 0, stream>>>(dinv);
    k_edgenorm<<<(N_EDGES + 255) / 256, 256, 0, stream>>>(src, dst, dinv, enorm);

    // ---- 5 GCN layers (finalize never re-reads X, so Xbuf can be both
    //      the input of layer L and the output of layer L)
    run_layer<32, 8>  (x,    W1, b1, Hbuf, Gbuf, Xbuf, src, dst, enorm, dinv, stream);
    run_layer<8, 16>  (Xbuf, W2, b2, Hbuf, Gbuf, Xbuf, src, dst, enorm, dinv, stream);
    run_layer<16, 64> (Xbuf, W3, b3, Hbuf, Gbuf, Xbuf, src, dst, enorm, dinv, stream);
    run_layer<64, 256>(Xbuf, W4, b4, Hbuf, Gbuf, Xbuf, src, dst, enorm, dinv, stream);
    run_layer<256, 512>(Xbuf, W5, b5, Hbuf, Gbuf, Xbuf, src, dst, enorm, dinv, stream);

    // ---- global mean pool
    k_fill<<<(N_GRAPHS * 512 + 255) / 256, 256, 0, stream>>>((float*)d_out, 0.0f, N_GRAPHS * 512);
    k_fill<<<(N_GRAPHS + 255) / 256, 256, 0, stream>>>(cnts, 0.0f, N_GRAPHS);
    k_counts<<<(N_NODES + 255) / 256, 256, 0, stream>>>(batch, cnts);
    k_poolsum<<<(N_NODES * 512 + 255) / 256, 256, 0, stream>>>(batch, Xbuf, (float*)d_out);
    k_pooldiv<<<(N_GRAPHS * 512 + 255) / 256, 256, 0, stream>>>((float*)d_out, cnts);
}